// RelativePositionEmbedding_20538533610132
// MI455X (gfx1250) — compile-verified
//
#include <hip/hip_runtime.h>
#include <hip/hip_bf16.h>

// out[d, k] = embedding[L-1, d]  for d in [0,32), k in [0,L)
// row_base = embedding + (L-1)*32 (host-computed).
//
// 2D grid: blockIdx.y = output row d (0..31)  -> no integer division,
//          blockIdx.x * blockDim.x + threadIdx.x = float4 column index.
// One global_store_b128 per thread; row value is a wave-uniform load.
__global__ void relpos_diag_bcast_v4(const float* __restrict__ row_base,
                                     float* __restrict__ out,
                                     int L4 /* L/4 */) {
    int d = blockIdx.y;                                  // output row
    int c = blockIdx.x * blockDim.x + threadIdx.x;       // float4 column
    if (c >= L4) return;
    float v = row_base[d];                               // uniform scalar load
    float4 vv = make_float4(v, v, v, v);
    reinterpret_cast<float4*>(out)[(size_t)d * L4 + c] = vv;  // b128 store
}

// Scalar fallback for L not divisible by 4 (not hit for L=2048).
__global__ void relpos_diag_bcast_s(const float* __restrict__ row_base,
                                    float* __restrict__ out,
                                    int L) {
    int d = blockIdx.y;
    int c = blockIdx.x * blockDim.x + threadIdx.x;
    if (c >= L) return;
    out[(size_t)d * L + c] = row_base[d];
}

extern "C" void kernel_launch(void* const* d_in, const int* in_sizes, int n_in,
                              void* d_out, int out_size, void* d_ws, size_t ws_size,
                              hipStream_t stream) {
    (void)in_sizes; (void)n_in; (void)d_ws; (void)ws_size;

    const float* emb = reinterpret_cast<const float*>(d_in[0]); // [2L-1, 32] f32
    float* out       = reinterpret_cast<float*>(d_out);         // [32, L] f32

    const int D = 32;
    const int L = out_size / D;                       // output is [32, seq_len]
    const float* row_base = emb + (size_t)(L - 1) * D;

    const int block = 256;                            // 8 wave32 waves
    if ((L & 3) == 0) {
        int L4 = L >> 2;                              // 512 float4s per row
        dim3 grid((L4 + block - 1) / block, D);       // (2, 32) for L=2048
        relpos_diag_bcast_v4<<<grid, block, 0, stream>>>(row_base, out, L4);
    } else {
        dim3 grid((L + block - 1) / block, D);
        relpos_diag_bcast_s<<<grid, block, 0, stream>>>(row_base, out, L);
    }
}